// LFQuantizer_19387482374538
// MI455X (gfx1250) — compile-verified
//
#include <hip/hip_runtime.h>

// Problem geometry (fixed by reference): z is (64, 13, 128, 128) f32
#define C_CH   13
#define HW     16384u            // 128*128
#define CHW    212992u           // 13*16384
#define NLOC   1048576u          // 64*128*128 spatial locations
#define NTOT   13631488u         // total elements
#define OUT_COMMIT 13631488u     // d_out offsets (flat f32, tuple order)
#define OUT_ENT    13631489u
#define OUT_IDX    13631490u

typedef __attribute__((ext_vector_type(2))) float v2f;
typedef __attribute__((ext_vector_type(8))) float v8f;

// Sum a value across all 32 lanes of a wave using V_WMMA_F32_16X16X4_F32.
// A-layout (ISA 7.12.2): lanes 0-15 supply A[m=lane, k=0..1] in VGPR0/1,
// lanes 16-31 supply A[m=lane-16, k=2..3].  With A[m,0]=v, A[m,1]=0 and
// B = all-ones (value identical in every slot, so B layout is irrelevant):
//   D[m,n] = v_m + v_{m+16}   for every n.
// D-layout: lane n<16 holds rows 0..7 of column n in its 8 VGPRs, lane n+16
// holds rows 8..15.  Per-lane sum of the 8 VGPRs + xor-16 partner = total.
__device__ __forceinline__ float wave_sum32_wmma(float v) {
    v2f a; a[0] = v;    a[1] = 0.0f;
    v2f b; b[0] = 1.0f; b[1] = 1.0f;
    v8f c = {};
    v8f d = __builtin_amdgcn_wmma_f32_16x16x4_f32(
        /*neg_a=*/false, a, /*neg_b=*/false, b,
        /*c_mod=*/(short)0, c, /*reuse_a=*/false, /*reuse_b=*/false);
    float s = ((d[0] + d[1]) + (d[2] + d[3])) + ((d[4] + d[5]) + (d[6] + d[7]));
    s += __shfl_xor(s, 16, 32);
    return s;
}

__global__ void lfq_zero_ws(float* ws) {
    if (threadIdx.x < 16) ws[threadIdx.x] = 0.0f;
}

__global__ __launch_bounds__(256) void lfq_main(const float* __restrict__ z,
                                                float* __restrict__ out,
                                                float* __restrict__ ws) {
    // Exact partition: 1024 blocks * 256 threads, one float4 location each.
    const unsigned tid  = blockIdx.x * 256u + threadIdx.x;
    const unsigned L    = tid * 4u;          // spatial location base (b*HW + hw)
    const unsigned b    = L >> 14;           // / HW
    const unsigned hw   = L & 16383u;        // % HW
    const unsigned base = b * CHW + hw;

    float pc[C_CH];
#pragma unroll
    for (int i = 0; i < C_CH; ++i) pc[i] = 0.0f;
    float sumsq = 0.0f, sument = 0.0f;
    int ibits[4] = {0, 0, 0, 0};

#pragma unroll
    for (int c = 0; c < C_CH; ++c) {
        if (c + 4 < C_CH)   // gfx1250 global_prefetch_b8, 4 channels ahead
            __builtin_prefetch(z + base + (unsigned)(c + 4) * HW, 0, 0);
        const unsigned off = base + (unsigned)c * HW;
        const float4 zv = *(const float4*)(z + off);
        const float zz[4] = {zv.x, zv.y, zv.z, zv.w};
        float qq[4];
        const int bit = 1 << (12 - c);
#pragma unroll
        for (int j = 0; j < 4; ++j) {
            const float zj = zz[j];
            const float qj = (zj > 0.0f) ? 1.0f : -1.0f;
            qq[j] = qj;
            if (zj > 0.0f) ibits[j] |= bit;
            const float dd = qj - zj;
            sumsq += dd * dd;
            // logits differ by t = 4z -> p0 = sigmoid(t); stable binary entropy:
            // H = ln(1+u) + a*u/(1+u),  u = exp(-|t|)
            const float t    = 4.0f * zj;
            const float aabs = fabsf(t);
            const float u    = __expf(-aabs);
            const float inv  = 1.0f / (1.0f + u);
            const float p0   = (t >= 0.0f) ? inv : u * inv;
            sument += __logf(1.0f + u) + aabs * (u * inv);
            pc[c]  += p0;
        }
        float4 qv; qv.x = qq[0]; qv.y = qq[1]; qv.z = qq[2]; qv.w = qq[3];
        *(float4*)(out + off) = qv;       // z_q4 = sign(z)
    }

    // bit-packed indices (MSB = channel 0), exact in f32
    {
        float* oi = out + OUT_IDX + L;    // region not 16B aligned -> scalar stores
        oi[0] = (float)ibits[0];
        oi[1] = (float)ibits[1];
        oi[2] = (float)ibits[2];
        oi[3] = (float)ibits[3];
    }

    // ---- block reduction of 15 statistics (13 channel p-sums, sumsq, sument)
    __shared__ float red[8][16];
    const int lane = (int)(threadIdx.x & 31u);
    const int wv   = (int)(threadIdx.x >> 5u);

    float stats[15];
#pragma unroll
    for (int i = 0; i < C_CH; ++i) stats[i] = pc[i];
    stats[13] = sumsq;
    stats[14] = sument;

#pragma unroll
    for (int i = 0; i < 15; ++i) {
        const float tot = wave_sum32_wmma(stats[i]);   // EXEC all-ones here
        if (lane == 0) red[wv][i] = tot;
    }
    __syncthreads();

    if (threadIdx.x < 15u) {
        float t = 0.0f;
#pragma unroll
        for (int w = 0; w < 8; ++w) t += red[w][threadIdx.x];
        atomicAdd(&ws[threadIdx.x], t);
    }
}

__global__ void lfq_finalize(const float* __restrict__ ws, float* __restrict__ out) {
    if (threadIdx.x == 0) {
        const float sumsq  = ws[13];
        const float sument = ws[14];
        // commit = 0.1 * (1 + 0.25) * mean((q-z)^2)
        const float commit = 0.1f * 1.25f * (sumsq * (1.0f / (float)NTOT));
        float me = 0.0f;
        for (int c = 0; c < C_CH; ++c) {
            float p0 = ws[c] * (1.0f / (float)NLOC);
            p0 = fminf(fmaxf(p0, 1e-12f), 1.0f - 1e-7f);
            const float p1 = 1.0f - p0;
            me -= p0 * __logf(p0) + p1 * __logf(p1);
        }
        me *= (1.0f / (float)C_CH);
        const float entv = 0.1f * (sument * (1.0f / (float)NTOT) - me);
        out[OUT_COMMIT] = commit;
        out[OUT_ENT]    = entv;
    }
}

extern "C" void kernel_launch(void* const* d_in, const int* in_sizes, int n_in,
                              void* d_out, int out_size, void* d_ws, size_t ws_size,
                              hipStream_t stream) {
    const float* z  = (const float*)d_in[0];
    float* out      = (float*)d_out;
    float* ws       = (float*)d_ws;

    lfq_zero_ws<<<1, 32, 0, stream>>>(ws);
    lfq_main<<<1024, 256, 0, stream>>>(z, out, ws);   // 1024*256*4 = NLOC
    lfq_finalize<<<1, 32, 0, stream>>>(ws, out);
}